// GMRNorm_27513560498872
// MI455X (gfx1250) — compile-verified
//
#include <hip/hip_runtime.h>
#include <hip/hip_bf16.h>

// GMRNorm for x[4, 8192, 1024] f32. Bandwidth-bound (268 MB min traffic ->
// ~11.5us at 23.3 TB/s). Three-phase: tile sums -> tile scan -> normalize,
// with phase 3 using CDNA5 async global->LDS copies (ASYNCcnt) to overlap the
// streaming reads with the barrier-synchronized per-timestep variance
// reductions, and non-temporal output stores to keep x resident in the 192MB
// L2 between phase 1 and phase 3.

#define B_  4
#define T_  8192
#define C_  1024
#define TT  128           // timesteps per tile
#define NT  (T_ / TT)     // 64 tiles
#define KT  4             // timesteps per async double-buffer stage (16 KB)
#define NSTAGE (TT / KT)  // 32 stages
#define NW  32            // waves per phase-3 block (1024 threads, wave32)
#define EPS_ 1e-5f

// ---- CDNA5 async global->LDS copy (tracked with ASYNCcnt) -------------------
// Probe result (round 1): builtin exists with 4 args and wants
// (v4i addrspace(1)*, v4i addrspace(3)*, imm offset, imm cpol).
typedef int v4i __attribute__((ext_vector_type(4)));
typedef __attribute__((address_space(1))) v4i* gv4i_p;
typedef __attribute__((address_space(3))) v4i* lv4i_p;

#if __has_builtin(__builtin_amdgcn_global_load_async_to_lds_b128)
#define ASYNC_B128(ldst, gsrc)                                                 \
  __builtin_amdgcn_global_load_async_to_lds_b128((gv4i_p)(gsrc),               \
                                                 (lv4i_p)(ldst), 0, 0)
#else
#define ASYNC_B128(ldst, gsrc)                                                 \
  asm volatile("global_load_async_to_lds_b128 %0, %1, off" ::                  \
               "v"((unsigned)(size_t)(__attribute__((address_space(3))) void*)(ldst)), \
               "v"((const void*)(gsrc))                                        \
               : "memory")
#endif

#if __has_builtin(__builtin_amdgcn_s_wait_asynccnt)
#define WAIT_ASYNC(n) __builtin_amdgcn_s_wait_asynccnt(n)
#else
#define WAIT_ASYNC(n) asm volatile("s_wait_asynccnt %0" ::"i"(n) : "memory")
#endif

// ---- Phase 1: per-tile channel sums ----------------------------------------
// grid = B*NT blocks, 256 threads; thread t covers channels [4t, 4t+4).
__global__ __launch_bounds__(256) void gmr_tilesum(const float4* __restrict__ x4,
                                                   float4* __restrict__ ts4) {
  const int tile = blockIdx.x % NT;
  const int b    = blockIdx.x / NT;
  const size_t base = (size_t)(b * T_ + tile * TT) * (C_ / 4) + threadIdx.x;
  float sx = 0.f, sy = 0.f, sz = 0.f, sw = 0.f;
  for (int r = 0; r < TT; ++r) {
    float4 v = x4[base + (size_t)r * (C_ / 4)];
    sx += v.x; sy += v.y; sz += v.z; sw += v.w;
  }
  ts4[(size_t)(b * NT + tile) * (C_ / 4) + threadIdx.x] =
      make_float4(sx, sy, sz, sw);
}

// ---- Phase 2: in-place exclusive scan over tiles per (b, channel) ----------
// grid = B blocks, 256 threads (float4 per thread). 1 MB total, trivial.
__global__ __launch_bounds__(256) void gmr_scan(float4* __restrict__ ts4) {
  const int b = blockIdx.x;
  float rx = 0.f, ry = 0.f, rz = 0.f, rw = 0.f;
  for (int tile = 0; tile < NT; ++tile) {
    const size_t idx = (size_t)(b * NT + tile) * (C_ / 4) + threadIdx.x;
    float4 v = ts4[idx];
    ts4[idx] = make_float4(rx, ry, rz, rw);
    rx += v.x; ry += v.y; rz += v.z; rw += v.w;
  }
}

// ---- Phase 3: prefix-mean + variance + normalize ---------------------------
// grid = B*NT blocks of 1024 threads (32 waves). Thread = one channel.
__global__ __launch_bounds__(1024) void gmr_norm(const float* __restrict__ x,
                                                 const float* __restrict__ weight,
                                                 const float* __restrict__ prefix,
                                                 float* __restrict__ out) {
  __shared__ __align__(16) float xbuf[2][KT * C_];  // 32 KB double buffer
  __shared__ __align__(16) float red[NW][KT];       // per-wave partial sums
  __shared__ float invb[KT];                        // rsqrt per timestep
  __shared__ float recip[TT];                       // 1/(t+1), 0 at t==0

  const int tid  = threadIdx.x;
  const int tile = blockIdx.x % NT;
  const int b    = blockIdx.x / NT;
  const int t0   = tile * TT;
  const int lane = tid & 31;
  const int wave = tid >> 5;

  if (tid < TT) {
    const int tg = t0 + tid;
    recip[tid] = (tg == 0) ? 0.0f : 1.0f / (float)(tg + 1);
  }

  float runsum  = prefix[(size_t)(b * NT + tile) * C_ + tid];
  const float w = weight[tid];
  const float* gbase = x + (size_t)(b * T_ + t0) * C_;

  // Prologue: stage 0 in flight (each thread copies 16B -> 16KB/stage total).
  ASYNC_B128(&xbuf[0][tid * 4], gbase + (size_t)tid * 4);

  for (int s = 0; s < NSTAGE; ++s) {
    if (s + 1 < NSTAGE) {
      // Issue next stage into the other buffer, then wait for current stage.
      ASYNC_B128(&xbuf[(s + 1) & 1][tid * 4],
                 gbase + (size_t)(s + 1) * KT * C_ + (size_t)tid * 4);
      WAIT_ASYNC(1);
    } else {
      WAIT_ASYNC(0);
    }
    __syncthreads();  // all waves' stage-s data landed in LDS (also recip)

    const float* xb = xbuf[s & 1];
    float xc[KT];
#pragma unroll
    for (int k = 0; k < KT; ++k) {
      const float xv = xb[k * C_ + tid];  // stride-1 across lanes: no conflicts
      runsum += xv;
      xc[k] = xv - runsum * recip[s * KT + k];  // mean==0 at global t==0
    }

    // Per-wave reduction of xc^2 over 32 lanes, 4 timesteps at once.
    float pk[KT];
#pragma unroll
    for (int k = 0; k < KT; ++k) {
      float p = xc[k] * xc[k];
#pragma unroll
      for (int off = 16; off > 0; off >>= 1) p += __shfl_xor(p, off, 32);
      pk[k] = p;
    }
    if (lane == 0)
      *(float4*)&red[wave][0] = make_float4(pk[0], pk[1], pk[2], pk[3]);
    __syncthreads();

    // Cross-wave reduction done by wave 0 (float4 LDS load, shuffle tree).
    if (wave == 0) {
      float4 v = *(const float4*)&red[lane][0];
#pragma unroll
      for (int off = 16; off > 0; off >>= 1) {
        v.x += __shfl_xor(v.x, off, 32);
        v.y += __shfl_xor(v.y, off, 32);
        v.z += __shfl_xor(v.z, off, 32);
        v.w += __shfl_xor(v.w, off, 32);
      }
      if (lane == 0) {
        invb[0] = rsqrtf(v.x * (1.0f / C_) + EPS_);
        invb[1] = rsqrtf(v.y * (1.0f / C_) + EPS_);
        invb[2] = rsqrtf(v.z * (1.0f / C_) + EPS_);
        invb[3] = rsqrtf(v.w * (1.0f / C_) + EPS_);
      }
    }
    __syncthreads();

    // Non-temporal stores: do not let 134MB of output evict x from L2.
#pragma unroll
    for (int k = 0; k < KT; ++k) {
      const int t = t0 + s * KT + k;
      __builtin_nontemporal_store(w * xc[k] * invb[k],
                                  out + (size_t)(b * T_ + t) * C_ + tid);
    }
  }
}

extern "C" void kernel_launch(void* const* d_in, const int* in_sizes, int n_in,
                              void* d_out, int out_size, void* d_ws, size_t ws_size,
                              hipStream_t stream) {
  const float* x      = (const float*)d_in[0];
  const float* weight = (const float*)d_in[1];
  float* out          = (float*)d_out;
  float* ts           = (float*)d_ws;  // B*NT*C floats = 1 MB scratch

  gmr_tilesum<<<B_ * NT, 256, 0, stream>>>((const float4*)x, (float4*)ts);
  gmr_scan<<<B_, 256, 0, stream>>>((float4*)ts);
  gmr_norm<<<B_ * NT, 1024, 0, stream>>>(x, weight, ts, out);
}